// FastEncoderBlock_71047349010918
// MI455X (gfx1250) — compile-verified
//
#include <hip/hip_runtime.h>

typedef __attribute__((ext_vector_type(2))) float v2f;
typedef __attribute__((ext_vector_type(8))) float v8f;

#define WMMA_F32(A,B,C) __builtin_amdgcn_wmma_f32_16x16x4_f32(false,(A),false,(B),(short)0,(C),false,false)

namespace {
constexpr int kB   = 4;
constexpr int kCIN = 32;
constexpr int kD   = 64;       // model dim / COUT
constexpr int kH   = 128;
constexpr int kW   = 128;
constexpr int kHW  = kH * kW;  // 16384
constexpr int kL   = kHW;      // sequence length per batch
constexpr int kRows = kB * kL; // 65536 total rows
constexpr int kN   = 16;       // d_state
constexpr int kNC  = 128;      // scan chunks per sequence
constexpr int kCL  = kL / kNC; // 128 steps per chunk
}

// ---------------------------------------------------------------------------
// Kernel 1: conv3x3 (as 9 shifted 1x1 WMMA GEMMs) + BN(eval) + ReLU
// One wave = 16 pixels (along W) x 64 out-channels.  Writes seq[b][l][c].
// ---------------------------------------------------------------------------
__global__ __launch_bounds__(256) void k_conv_stem(
    const float* __restrict__ x, const float* __restrict__ conv_w,
    const float* __restrict__ gam, const float* __restrict__ bet,
    const float* __restrict__ mean_, const float* __restrict__ var_,
    float* __restrict__ seq)
{
  // per-tap weights staged in LDS: [kpair(16)][cout(64)][2] = 2048 floats = 8KB
  __shared__ __align__(16) float wl[kCIN / 2 * kD * 2];

  const int wave = threadIdx.x >> 5;
  const int lane = threadIdx.x & 31;
  const int tile = blockIdx.x * 8 + wave;       // 4096 tiles total
  const int b  = tile >> 10;                    // 1024 tiles per batch
  const int hw = tile & 1023;
  const int h  = hw >> 3;
  const int w0 = (hw & 7) << 4;

  const int m   = lane & 15;        // pixel within tile (M)
  const int n   = lane & 15;        // out-channel within N-tile
  const int kh2 = (lane >> 4) << 1; // K offset: 0 or 2
  const int mbase = (lane >> 4) * 8;
  const int w_pix = w0 + m;

  v8f a0 = {}, a1 = {}, a2 = {}, a3 = {};

  for (int t = 0; t < 9; ++t) {
    __syncthreads();
    // stage tap t weights: conv_w[(o*32+c)*9+t] -> wl[((c>>1)*64+o)*2+(c&1)]
    for (int i = threadIdx.x; i < kCIN * kD; i += 256) {
      int o = i >> 5; int c = i & 31;
      wl[(((c >> 1) * kD) + o) * 2 + (c & 1)] = conv_w[(o * kCIN + c) * 9 + t];
    }
    __syncthreads();

    const int dy = t / 3 - 1, dx = t % 3 - 1;
    const int hy = h + dy;
    const int wx = w_pix + dx;
    const bool val = (hy >= 0) && (hy < kH) && (wx >= 0) && (wx < kW);
    const float* xb = x + b * kCIN * kHW + (val ? hy * kW + wx : 0);

#pragma unroll
    for (int k0 = 0; k0 < kCIN; k0 += 4) {
      const int kk = k0 + kh2;
      float t0 = xb[kk * kHW];
      float t1 = xb[(kk + 1) * kHW];
      v2f a; a.x = val ? t0 : 0.f; a.y = val ? t1 : 0.f;
      const float* wp2 = wl + (kk >> 1) * kD * 2;
      v2f b0 = *(const v2f*)(wp2 + (n      ) * 2);
      v2f b1 = *(const v2f*)(wp2 + (n + 16 ) * 2);
      v2f b2 = *(const v2f*)(wp2 + (n + 32 ) * 2);
      v2f b3 = *(const v2f*)(wp2 + (n + 48 ) * 2);
      a0 = WMMA_F32(a, b0, a0);
      a1 = WMMA_F32(a, b1, a1);
      a2 = WMMA_F32(a, b2, a2);
      a3 = WMMA_F32(a, b3, a3);
    }
  }

  // BN + ReLU epilogue; element (M=mbase+r, N=nt*16+n)
  const int lrow0 = h * kW + w0;
  float* sb = seq + b * kL * kD;
#define BNST(ACC, NT)                                                     \
  { int c = (NT) * 16 + n;                                                \
    float inv = gam[c] * rsqrtf(var_[c] + 1e-5f);                         \
    float off = bet[c] - mean_[c] * inv;                                  \
    _Pragma("unroll") for (int r = 0; r < 8; ++r) {                       \
      float vv = (ACC)[r] * inv + off;                                    \
      vv = vv > 0.f ? vv : 0.f;                                           \
      sb[(lrow0 + mbase + r) * kD + c] = vv; } }
  BNST(a0, 0) BNST(a1, 1) BNST(a2, 2) BNST(a3, 3)
#undef BNST
}

// ---------------------------------------------------------------------------
// Kernel 2: LayerNorm over d=64.  16 lanes per row, 2 rows per wave.
// ---------------------------------------------------------------------------
__global__ __launch_bounds__(256) void k_ln(
    const float* __restrict__ seq, const float* __restrict__ gam,
    const float* __restrict__ bet, float* __restrict__ out)
{
  const int gt = blockIdx.x * 256 + threadIdx.x;
  const int row = gt >> 4, l16 = gt & 15;
  const float* r = seq + row * kD;
  float v0 = r[l16], v1 = r[l16 + 16], v2 = r[l16 + 32], v3 = r[l16 + 48];
  float s  = v0 + v1 + v2 + v3;
  float ss = v0 * v0 + v1 * v1 + v2 * v2 + v3 * v3;
#pragma unroll
  for (int o = 8; o >= 1; o >>= 1) {
    s  += __shfl_xor(s, o, 32);
    ss += __shfl_xor(ss, o, 32);
  }
  const float mu = s * (1.f / 64.f);
  const float rstd = rsqrtf(ss * (1.f / 64.f) - mu * mu + 1e-5f);
  float* w = out + row * kD;
  w[l16]      = (v0 - mu) * rstd * gam[l16]      + bet[l16];
  w[l16 + 16] = (v1 - mu) * rstd * gam[l16 + 16] + bet[l16 + 16];
  w[l16 + 32] = (v2 - mu) * rstd * gam[l16 + 32] + bet[l16 + 32];
  w[l16 + 48] = (v3 - mu) * rstd * gam[l16 + 48] + bet[l16 + 48];
}

// ---------------------------------------------------------------------------
// Kernel 3: in_proj GEMM [rows x 64] @ W^T (128x64) -> xi_raw / z halves.
// One wave = 16 rows x 64 cols of one half.
// ---------------------------------------------------------------------------
__global__ __launch_bounds__(256) void k_inproj(
    const float* __restrict__ h, const float* __restrict__ wp,
    float* __restrict__ xi_raw, float* __restrict__ zb)
{
  const int wave = threadIdx.x >> 5, lane = threadIdx.x & 31;
  const int job = blockIdx.x * 8 + wave;   // 8192 jobs
  const int mtile = job >> 1, half = job & 1;
  const int row0 = mtile * 16;
  const int m = lane & 15, n = lane & 15;
  const int kh = (lane >> 4) << 1, mbase = (lane >> 4) * 8;
  const float* wb = wp + half * 64 * kD;
  v8f a0 = {}, a1 = {}, a2 = {}, a3 = {};
#pragma unroll
  for (int k0 = 0; k0 < kD; k0 += 4) {
    const int kk = k0 + kh;
    v2f a  = *(const v2f*)(h  + (row0 + m) * kD + kk);
    v2f b0 = *(const v2f*)(wb + (n      ) * kD + kk);
    v2f b1 = *(const v2f*)(wb + (n + 16 ) * kD + kk);
    v2f b2 = *(const v2f*)(wb + (n + 32 ) * kD + kk);
    v2f b3 = *(const v2f*)(wb + (n + 48 ) * kD + kk);
    a0 = WMMA_F32(a, b0, a0);
    a1 = WMMA_F32(a, b1, a1);
    a2 = WMMA_F32(a, b2, a2);
    a3 = WMMA_F32(a, b3, a3);
  }
  float* dst = half ? zb : xi_raw;
#define INPST(ACC, NT)                                                    \
  { _Pragma("unroll") for (int r = 0; r < 8; ++r)                         \
      dst[(row0 + mbase + r) * kD + (NT) * 16 + n] = (ACC)[r]; }
  INPST(a0, 0) INPST(a1, 1) INPST(a2, 2) INPST(a3, 3)
#undef INPST
}

// ---------------------------------------------------------------------------
// Kernel 4: depthwise causal conv1d (K=3, left pad 2) + bias + SiLU.
// float4 over channels.
// ---------------------------------------------------------------------------
__global__ __launch_bounds__(256) void k_dwconv(
    const float* __restrict__ xr, const float* __restrict__ cw,
    const float* __restrict__ bias, float* __restrict__ xo)
{
  const int i  = blockIdx.x * 256 + threadIdx.x;  // over kRows*16 float4s
  const int c4 = i & 15;
  const int row = i >> 4;
  const int l = row & (kL - 1);
  const float4* cur = ((const float4*)xr) + row * 16 + c4;
  const float4 z4 = make_float4(0.f, 0.f, 0.f, 0.f);
  float4 v  = cur[0];
  float4 p1 = (l >= 1) ? cur[-16] : z4;
  float4 p2 = (l >= 2) ? cur[-32] : z4;
  const int c = c4 * 4;
  float4 o;
#define DW(comp, ci)                                                      \
  { float t = cw[(ci) * 3 + 0] * p2.comp + cw[(ci) * 3 + 1] * p1.comp +   \
              cw[(ci) * 3 + 2] * v.comp + bias[(ci)];                     \
    o.comp = t / (1.f + __expf(-t)); }
  DW(x, c + 0) DW(y, c + 1) DW(z, c + 2) DW(w, c + 3)
#undef DW
  ((float4*)xo)[row * 16 + c4] = o;
}

// ---------------------------------------------------------------------------
// Kernel 5: x_proj GEMM (N padded 36->48) + dt_proj (single K=4 WMMA per
// N-tile via LDS relayout) + softplus.  Emits delta, Bm, Cm.
// ---------------------------------------------------------------------------
__global__ __launch_bounds__(256) void k_xproj(
    const float* __restrict__ u, const float* __restrict__ xw,
    const float* __restrict__ dtw, const float* __restrict__ dtb,
    float* __restrict__ delta, float* __restrict__ Bm, float* __restrict__ Cm)
{
  __shared__ float dtl[8][64];  // per-wave 16x4 dt tile
  const int wave = threadIdx.x >> 5, lane = threadIdx.x & 31;
  const int mtile = blockIdx.x * 8 + wave;  // 4096
  const int row0 = mtile * 16;
  const int m = lane & 15, n = lane & 15;
  const int kh = (lane >> 4) << 1, mbase = (lane >> 4) * 8;
  const bool ok2 = (32 + n) < 36;
  v8f a0 = {}, a1 = {}, a2 = {};
#pragma unroll
  for (int k0 = 0; k0 < kD; k0 += 4) {
    const int kk = k0 + kh;
    v2f a  = *(const v2f*)(u  + (row0 + m) * kD + kk);
    v2f b0 = *(const v2f*)(xw + (n      ) * kD + kk);
    v2f b1 = *(const v2f*)(xw + (n + 16 ) * kD + kk);
    v2f b2;
    if (ok2) b2 = *(const v2f*)(xw + (n + 32) * kD + kk);
    else { b2.x = 0.f; b2.y = 0.f; }
    a0 = WMMA_F32(a, b0, a0);
    a1 = WMMA_F32(a, b1, a1);
    a2 = WMMA_F32(a, b2, a2);
  }
  // scatter: cols 0..3 -> dt (LDS), 4..19 -> Bm, 20..35 -> Cm
#pragma unroll
  for (int r = 0; r < 8; ++r) {
    const int mr = mbase + r;
    const int rr = row0 + mr;
    float v0v = a0[r];
    if (n < 4) dtl[wave][mr * 4 + n] = v0v;
    else       Bm[rr * 16 + (n - 4)] = v0v;
    float v1v = a1[r];
    if (n < 4) Bm[rr * 16 + 12 + n] = v1v;
    else       Cm[rr * 16 + (n - 4)] = v1v;
    if (n < 4) Cm[rr * 16 + 12 + n] = a2[r];
  }
  __syncthreads();
  // dt_proj: [16x4] @ [4x64] -> one WMMA per N-tile
  v2f ad; ad.x = dtl[wave][m * 4 + kh]; ad.y = dtl[wave][m * 4 + kh + 1];
  v2f bd0 = *(const v2f*)(dtw + (n      ) * 4 + kh);
  v2f bd1 = *(const v2f*)(dtw + (n + 16 ) * 4 + kh);
  v2f bd2 = *(const v2f*)(dtw + (n + 32 ) * 4 + kh);
  v2f bd3 = *(const v2f*)(dtw + (n + 48 ) * 4 + kh);
  v8f d0 = {}, d1 = {}, d2 = {}, d3 = {};
  d0 = WMMA_F32(ad, bd0, d0);
  d1 = WMMA_F32(ad, bd1, d1);
  d2 = WMMA_F32(ad, bd2, d2);
  d3 = WMMA_F32(ad, bd3, d3);
#define SPST(ACC, NT)                                                     \
  { _Pragma("unroll") for (int r = 0; r < 8; ++r) {                       \
      int c = (NT) * 16 + n;                                              \
      float vv = (ACC)[r] + dtb[c];                                       \
      float sp = vv > 20.f ? vv : log1pf(__expf(vv));                     \
      delta[(row0 + mbase + r) * kD + c] = sp; } }
  SPST(d0, 0) SPST(d1, 1) SPST(d2, 2) SPST(d3, 3)
#undef SPST
}

// ---------------------------------------------------------------------------
// Chunked selective scan.  lane = channel d (32 channels per wave),
// h[16] states in registers, B/C rows are wave-uniform float4 broadcasts.
// job bits: dhalf(1) | chunk(7) | dir(1) | b(2)
// ---------------------------------------------------------------------------
__device__ __forceinline__ void scan_decode(int job, int lane,
    int& b, int& dir, int& chunk, int& d)
{
  int dh = job & 1; int t = job >> 1;
  chunk = t & (kNC - 1); t >>= 7;
  dir = t & 1; b = t >> 1;
  d = dh * 32 + lane;
}

__global__ __launch_bounds__(256) void k_scanA(
    const float* __restrict__ delta, const float* __restrict__ u,
    const float* __restrict__ Bm, const float* __restrict__ A_log,
    float* __restrict__ P, float* __restrict__ hend)
{
  const int wave = threadIdx.x >> 5, lane = threadIdx.x & 31;
  int b, dir, chunk, d;
  scan_decode(blockIdx.x * 8 + wave, lane, b, dir, chunk, d);
  float a[kN], h[kN], p[kN];
#pragma unroll
  for (int i = 0; i < kN; ++i) {
    a[i] = -__expf(A_log[d * kN + i]); h[i] = 0.f; p[i] = 1.f;
  }
  const int base = b * kL;
  int l = dir ? (kL - 1 - chunk * kCL) : chunk * kCL;
  const int stp = dir ? -1 : 1;
  for (int s = 0; s < kCL; ++s, l += stp) {
    if (s + 4 < kCL) {
      __builtin_prefetch(delta + (base + l + 4 * stp) * kD + d, 0, 1);
      __builtin_prefetch(Bm + (base + l + 4 * stp) * 16, 0, 1);
    }
    const int rr = base + l;
    const float dl = delta[rr * kD + d];
    const float uu = u[rr * kD + d];
    const float du = dl * uu;
    const float4* bp = (const float4*)(Bm + rr * 16);
    float4 B0 = bp[0], B1 = bp[1], B2 = bp[2], B3 = bp[3];
    const float bv[kN] = {B0.x, B0.y, B0.z, B0.w, B1.x, B1.y, B1.z, B1.w,
                          B2.x, B2.y, B2.z, B2.w, B3.x, B3.y, B3.z, B3.w};
#pragma unroll
    for (int i = 0; i < kN; ++i) {
      float dA = __expf(dl * a[i]);
      h[i] = dA * h[i] + du * bv[i];
      p[i] *= dA;
    }
  }
  const int chain = ((b * 2 + dir) * kD + d) * kN;
#pragma unroll
  for (int i = 0; i < kN; ++i) {
    P[(chain + i) * kNC + chunk]    = p[i];
    hend[(chain + i) * kNC + chunk] = h[i];
  }
}

__global__ __launch_bounds__(256) void k_scanB(
    const float* __restrict__ P, const float* __restrict__ hend,
    float* __restrict__ Hin)
{
  const int chain = blockIdx.x * 256 + threadIdx.x;  // 8192 chains
  const float* Pp = P + chain * kNC;
  const float* he = hend + chain * kNC;
  float* Hi = Hin + chain * kNC;
  float Hv = 0.f;
  for (int j = 0; j < kNC; ++j) {
    Hi[j] = Hv;
    Hv = Pp[j] * Hv + he[j];
  }
}

__global__ __launch_bounds__(256) void k_scanC(
    const float* __restrict__ delta, const float* __restrict__ u,
    const float* __restrict__ Bm, const float* __restrict__ Cm,
    const float* __restrict__ A_log, const float* __restrict__ Dv,
    const float* __restrict__ Hin, float* __restrict__ yf,
    float* __restrict__ yb)
{
  const int wave = threadIdx.x >> 5, lane = threadIdx.x & 31;
  int b, dir, chunk, d;
  scan_decode(blockIdx.x * 8 + wave, lane, b, dir, chunk, d);
  const float Dd = Dv[d];
  const int chain = ((b * 2 + dir) * kD + d) * kN;
  float a[kN], h[kN];
#pragma unroll
  for (int i = 0; i < kN; ++i) {
    a[i] = -__expf(A_log[d * kN + i]);
    h[i] = Hin[(chain + i) * kNC + chunk];
  }
  float* yo = dir ? yb : yf;
  const int base = b * kL;
  int l = dir ? (kL - 1 - chunk * kCL) : chunk * kCL;
  const int stp = dir ? -1 : 1;
  for (int s = 0; s < kCL; ++s, l += stp) {
    if (s + 4 < kCL) {
      __builtin_prefetch(delta + (base + l + 4 * stp) * kD + d, 0, 1);
      __builtin_prefetch(Bm + (base + l + 4 * stp) * 16, 0, 1);
      __builtin_prefetch(Cm + (base + l + 4 * stp) * 16, 0, 1);
    }
    const int rr = base + l;
    const float dl = delta[rr * kD + d];
    const float uu = u[rr * kD + d];
    const float du = dl * uu;
    const float4* bp = (const float4*)(Bm + rr * 16);
    const float4* cp = (const float4*)(Cm + rr * 16);
    float4 B0 = bp[0], B1 = bp[1], B2 = bp[2], B3 = bp[3];
    float4 C0 = cp[0], C1 = cp[1], C2 = cp[2], C3 = cp[3];
    const float bv[kN] = {B0.x, B0.y, B0.z, B0.w, B1.x, B1.y, B1.z, B1.w,
                          B2.x, B2.y, B2.z, B2.w, B3.x, B3.y, B3.z, B3.w};
    const float cv[kN] = {C0.x, C0.y, C0.z, C0.w, C1.x, C1.y, C1.z, C1.w,
                          C2.x, C2.y, C2.z, C2.w, C3.x, C3.y, C3.z, C3.w};
    float y = 0.f;
#pragma unroll
    for (int i = 0; i < kN; ++i) {
      float dA = __expf(dl * a[i]);
      h[i] = dA * h[i] + du * bv[i];
      y += h[i] * cv[i];
    }
    yo[rr * kD + d] = y + uu * Dd;
  }
}

// ---------------------------------------------------------------------------
// Kernel: g = (y_f + y_b) * silu(z), written in place over z.
// ---------------------------------------------------------------------------
__global__ __launch_bounds__(256) void k_gate(
    const float* __restrict__ yf, const float* __restrict__ yb,
    float* __restrict__ z)
{
  const int i = blockIdx.x * 256 + threadIdx.x;  // 1,048,576 float4s
  float4 f = ((const float4*)yf)[i];
  float4 g = ((const float4*)yb)[i];
  float4 zz = ((float4*)z)[i];
  float4 o;
  o.x = (f.x + g.x) * (zz.x / (1.f + __expf(-zz.x)));
  o.y = (f.y + g.y) * (zz.y / (1.f + __expf(-zz.y)));
  o.z = (f.z + g.z) * (zz.z / (1.f + __expf(-zz.z)));
  o.w = (f.w + g.w) * (zz.w / (1.f + __expf(-zz.w)));
  ((float4*)z)[i] = o;
}

// ---------------------------------------------------------------------------
// Kernel: out = g @ out_proj_w^T + seq, then NCHW transpose-store via LDS.
// ---------------------------------------------------------------------------
__global__ __launch_bounds__(256) void k_outproj(
    const float* __restrict__ g, const float* __restrict__ w,
    const float* __restrict__ seq, float* __restrict__ out)
{
  __shared__ __align__(16) float tile[8][kD * 16];
  const int wave = threadIdx.x >> 5, lane = threadIdx.x & 31;
  const int mtile = blockIdx.x * 8 + wave;  // 4096
  const int row0 = mtile * 16;
  const int m = lane & 15, n = lane & 15;
  const int kh = (lane >> 4) << 1, mbase = (lane >> 4) * 8;
  v8f a0 = {}, a1 = {}, a2 = {}, a3 = {};
#pragma unroll
  for (int k0 = 0; k0 < kD; k0 += 4) {
    const int kk = k0 + kh;
    v2f a  = *(const v2f*)(g + (row0 + m) * kD + kk);
    v2f b0 = *(const v2f*)(w + (n      ) * kD + kk);
    v2f b1 = *(const v2f*)(w + (n + 16 ) * kD + kk);
    v2f b2 = *(const v2f*)(w + (n + 32 ) * kD + kk);
    v2f b3 = *(const v2f*)(w + (n + 48 ) * kD + kk);
    a0 = WMMA_F32(a, b0, a0);
    a1 = WMMA_F32(a, b1, a1);
    a2 = WMMA_F32(a, b2, a2);
    a3 = WMMA_F32(a, b3, a3);
  }
#define K9ST(ACC, NT)                                                     \
  { _Pragma("unroll") for (int r = 0; r < 8; ++r) {                       \
      int mr = mbase + r; int c = (NT) * 16 + n;                          \
      tile[wave][c * 16 + mr] = (ACC)[r] + seq[(row0 + mr) * kD + c]; } }
  K9ST(a0, 0) K9ST(a1, 1) K9ST(a2, 2) K9ST(a3, 3)
#undef K9ST
  __syncthreads();
  const int b   = row0 / kL;
  const int l0p = row0 & (kL - 1);
#pragma unroll
  for (int q = 0; q < 2; ++q) {
    const int c = lane * 2 + q;
    float4* dst = (float4*)(out + (b * kD + c) * kL + l0p);
    const float4* src = (const float4*)(&tile[wave][c * 16]);
    dst[0] = src[0]; dst[1] = src[1]; dst[2] = src[2]; dst[3] = src[3];
  }
}

// ---------------------------------------------------------------------------
extern "C" void kernel_launch(void* const* d_in, const int* in_sizes, int n_in,
                              void* d_out, int out_size, void* d_ws, size_t ws_size,
                              hipStream_t stream)
{
  const float* x      = (const float*)d_in[0];
  const float* conv_w = (const float*)d_in[1];
  const float* bn_g   = (const float*)d_in[2];
  const float* bn_b   = (const float*)d_in[3];
  const float* bn_m   = (const float*)d_in[4];
  const float* bn_v   = (const float*)d_in[5];
  const float* ln_g   = (const float*)d_in[6];
  const float* ln_b   = (const float*)d_in[7];
  const float* inw    = (const float*)d_in[8];
  const float* c1w    = (const float*)d_in[9];
  const float* c1b    = (const float*)d_in[10];
  const float* xpw    = (const float*)d_in[11];
  const float* dtw    = (const float*)d_in[12];
  const float* dtb    = (const float*)d_in[13];
  const float* A_log  = (const float*)d_in[14];
  const float* Dv     = (const float*)d_in[15];
  const float* outw   = (const float*)d_in[16];
  float* out = (float*)d_out;

  float* ws = (float*)d_ws;
  const size_t R64 = (size_t)kRows * kD;        // 4,194,304 floats
  const size_t R16 = (size_t)kRows * 16;        // 1,048,576 floats
  const size_t CH  = (size_t)8 * kD * kN * kNC; // 1,048,576 floats
  float* seq   = ws;
  float* hln   = seq + R64;      // reused as y_f after in_proj
  float* xiraw = hln + R64;      // reused as y_b after dwconv
  float* zb    = xiraw + R64;    // reused as g after gating
  float* xi    = zb + R64;
  float* dl    = xi + R64;
  float* Bmb   = dl + R64;
  float* Cmb   = Bmb + R16;
  float* Pb    = Cmb + R16;
  float* heb   = Pb + CH;
  float* Hib   = heb + CH;

  k_conv_stem<<<512, 256, 0, stream>>>(x, conv_w, bn_g, bn_b, bn_m, bn_v, seq);
  k_ln<<<4096, 256, 0, stream>>>(seq, ln_g, ln_b, hln);
  k_inproj<<<1024, 256, 0, stream>>>(hln, inw, xiraw, zb);
  k_dwconv<<<4096, 256, 0, stream>>>(xiraw, c1w, c1b, xi);
  k_xproj<<<512, 256, 0, stream>>>(xi, xpw, dtw, dtb, dl, Bmb, Cmb);
  k_scanA<<<256, 256, 0, stream>>>(dl, xi, Bmb, A_log, Pb, heb);
  k_scanB<<<32, 256, 0, stream>>>(Pb, heb, Hib);
  float* yf = hln;
  float* yb = xiraw;
  k_scanC<<<256, 256, 0, stream>>>(dl, xi, Bmb, Cmb, A_log, Dv, Hib, yf, yb);
  k_gate<<<4096, 256, 0, stream>>>(yf, yb, zb);
  k_outproj<<<512, 256, 0, stream>>>(zb, outw, seq, out);
}